// QGINLayer_54228257079522
// MI455X (gfx1250) — compile-verified
//
#include <hip/hip_runtime.h>
#include <math.h>

typedef float v2f __attribute__((ext_vector_type(2)));
typedef float v8f __attribute__((ext_vector_type(8)));
typedef int   gi4 __attribute__((vector_size(16)));   // matches async-LDS builtin param
typedef __attribute__((address_space(1))) gi4* gptr4; // global int4*
typedef __attribute__((address_space(3))) gi4* lptr4; // LDS int4*

#define IN_F 256
// K-pair interleaved Hamilton: Hp[p][n] = {H[2p][n], H[2p+1][n]}, p=0..127, n=0..255
// global row = 512 floats (2048B); LDS row padded to 544 floats (2176B == 32 banks mod 64)
#define LDSP 544

// ---------------------------------------------------------------- zero fill
__global__ __launch_bounds__(256) void zero_f32(float* __restrict__ p, size_t n) {
    size_t i = (size_t)blockIdx.x * blockDim.x + threadIdx.x;
    if (i < n) p[i] = 0.0f;
}

// ------------------------------------------------- hamilton (quaternion) build
// hamilton[row,col] = sign(br,bc) * w[row%64, comp(br,bc)*64 + col%64]
// stored K-pair interleaved: Hp[(row>>1)*512 + col*2 + (row&1)]
__global__ __launch_bounds__(256) void build_hamilton(const float* __restrict__ w,
                                                      float* __restrict__ Hp) {
    const int idx = blockIdx.x * 256 + threadIdx.x;       // 65536 threads
    const int row = idx >> 8, col = idx & 255;
    const int br = row >> 6, rr = row & 63;
    const int bc = col >> 6, cc = col & 63;
    const int   compT[16] = {0,1,2,3,  1,0,3,2,  2,3,0,1,  3,2,1,0};
    const float signT[16] = {1,1,1,1, -1,1,1,-1, -1,-1,1,1, -1,1,-1,1};
    const int t = br * 4 + bc;
    const float val = signT[t] * w[rr * 256 + compT[t] * 64 + cc];
    Hp[(row >> 1) * 512 + col * 2 + (row & 1)] = val;
}

// ------------------------------------------------------------------ SpMM
// one thread = (edge, 4-feature group); 64 consecutive threads share one edge
__global__ __launch_bounds__(256) void spmm_atomic(const float* __restrict__ x,
                                                   const int*   __restrict__ rows,
                                                   const int*   __restrict__ cols,
                                                   const float* __restrict__ ev,
                                                   float* __restrict__ acc,
                                                   int nEdges) {
    const long long idx = (long long)blockIdx.x * blockDim.x + threadIdx.x;
    const int e = (int)(idx >> 6);
    if (e >= nEdges) return;
    const int f = ((int)idx & 63) << 2;
    const float v = ev[e];
    const int c = cols[e];
    const int r = rows[e];
    const float4 xv = *(const float4*)(x + (size_t)c * IN_F + f);
    float* dst = acc + (size_t)r * IN_F + f;
    unsafeAtomicAdd(dst + 0, v * xv.x);
    unsafeAtomicAdd(dst + 1, v * xv.y);
    unsafeAtomicAdd(dst + 2, v * xv.z);
    unsafeAtomicAdd(dst + 3, v * xv.w);
}

// ------------------------------------------------------------- fp32 WMMA GEMM
// out[M,256] = A[M,256] @ H[256,256], H given K-pair interleaved (Hp).
// Hp staged whole in LDS (272KB < 320KB WGP LDS) via async global->LDS B128 copies.
// One wave -> 16(M) x 64(N): 4 accumulator tiles sharing each A fragment (4x A reuse).
// V_WMMA_F32_16X16X4_F32 fragment layout:
//   A (16x4): lanes 0-15 hold K=k,k+1 ; lanes 16-31 hold K=k+2,k+3 ; M = lane&15
//   B (4x16): same K split; N = lane&15  -> one ds_load_b64 of Hp pair per fragment
//   C/D     : VGPR v -> M = (lane>>4)*8 + v, N = lane&15
__global__ __launch_bounds__(256) void gemm256_wmma(const float* __restrict__ A,
                                                    const float* __restrict__ Hp,
                                                    float* __restrict__ out,
                                                    int mTiles) {
    __shared__ float Hs[128 * LDSP];
    const int tid = threadIdx.x;
#if __has_builtin(__builtin_amdgcn_global_load_async_to_lds_b128)
    for (int i = tid; i < 128 * 128; i += 256) {          // 16384 x 16B chunks
        const int p = i >> 7, c4 = (i & 127) << 2;
        __builtin_amdgcn_global_load_async_to_lds_b128(
            (gptr4)(Hp + p * 512 + c4),
            (lptr4)(&Hs[p * LDSP + c4]),
            0, 0);
    }
#if __has_builtin(__builtin_amdgcn_s_wait_asynccnt)
    __builtin_amdgcn_s_wait_asynccnt(0);
#else
    asm volatile("s_wait_asynccnt 0x0" ::: "memory");
#endif
#else
    for (int i = tid; i < 128 * 128; i += 256) {
        const int p = i >> 7, c4 = (i & 127) << 2;
        *(float4*)(&Hs[p * LDSP + c4]) = *(const float4*)(Hp + p * 512 + c4);
    }
#endif
    __syncthreads();

    const int wave = tid >> 5, lane = tid & 31;
    const int wgid = blockIdx.x * 8 + wave;     // global wave index
    const int mTile = wgid >> 2;                // one M tile per wave
    const int nGroup = wgid & 3;                // 64-wide N group per wave
    if (mTile >= mTiles) return;

    const int laneM = lane & 15;
    const int halfK = (lane >> 4) << 1;         // 0 or 2  (A k offset)
    const int halfP = (lane >> 4);              // 0 or 1  (B pair offset)
    const float* Arow  = A + (size_t)(mTile * 16 + laneM) * 256;
    const float* Hbase = Hs + (nGroup * 64 + laneM) * 2;

    v8f acc0 = {}, acc1 = {}, acc2 = {}, acc3 = {};
#pragma unroll 4
    for (int k = 0; k < 256; k += 4) {
        const v2f a = *(const v2f*)(Arow + k + halfK);
        const float* hrow = Hbase + ((k >> 1) + halfP) * LDSP;
        const v2f b0 = *(const v2f*)(hrow +  0);   // n + 0
        const v2f b1 = *(const v2f*)(hrow + 32);   // n + 16
        const v2f b2 = *(const v2f*)(hrow + 64);   // n + 32
        const v2f b3 = *(const v2f*)(hrow + 96);   // n + 48
        acc0 = __builtin_amdgcn_wmma_f32_16x16x4_f32(false, a, false, b0, (short)0, acc0, false, false);
        acc1 = __builtin_amdgcn_wmma_f32_16x16x4_f32(false, a, false, b1, (short)0, acc1, false, false);
        acc2 = __builtin_amdgcn_wmma_f32_16x16x4_f32(false, a, false, b2, (short)0, acc2, false, false);
        acc3 = __builtin_amdgcn_wmma_f32_16x16x4_f32(false, a, false, b3, (short)0, acc3, false, false);
    }

    float* orow = out + (size_t)(mTile * 16 + ((lane >> 4) << 3)) * 256
                + nGroup * 64 + laneM;
#pragma unroll
    for (int v = 0; v < 8; ++v) {
        orow[(size_t)v * 256 +  0] = acc0[v];
        orow[(size_t)v * 256 + 16] = acc1[v];
        orow[(size_t)v * 256 + 32] = acc2[v];
        orow[(size_t)v * 256 + 48] = acc3[v];
    }
}

// -------------------------------------------------------------- BN statistics
// block = 256 threads (one per channel), each block covers 128 rows
__global__ __launch_bounds__(256) void bn_stats(const float* __restrict__ y,
                                                float* __restrict__ sums,
                                                float* __restrict__ sumsq,
                                                int nRows) {
    const int c = threadIdx.x;
    const int rowBase = blockIdx.x * 128;
    const int rEnd = min(128, nRows - rowBase);
    const float* p = y + (size_t)rowBase * 256 + c;
    float s = 0.0f, s2 = 0.0f;
    for (int r = 0; r < rEnd; ++r) {
        const float v = p[(size_t)r * 256];
        s += v; s2 += v * v;
    }
    unsafeAtomicAdd(&sums[c], s);
    unsafeAtomicAdd(&sumsq[c], s2);
}

__global__ __launch_bounds__(256) void bn_finalize(const float* __restrict__ sums,
                                                   const float* __restrict__ sumsq,
                                                   const float* __restrict__ gamma,
                                                   const float* __restrict__ beta,
                                                   float* __restrict__ scsh,
                                                   int nRows) {
    const int c = threadIdx.x;
    const float inv = 1.0f / (float)nRows;
    const float mean = sums[c] * inv;
    const float var = sumsq[c] * inv - mean * mean;
    const float sc = gamma[c] * rsqrtf(var + 1e-5f);
    scsh[c]       = sc;
    scsh[256 + c] = beta[c] - mean * sc;
}

// ------------------------------------------------ in-place BN scale/shift + tanh
__global__ __launch_bounds__(256) void bn_tanh_apply(float* __restrict__ y,
                                                     const float* __restrict__ scsh,
                                                     size_t n4) {
    const size_t i = (size_t)blockIdx.x * blockDim.x + threadIdx.x;
    if (i >= n4) return;
    const int f = ((int)i & 63) << 2;                 // channel of first element
    float4 v = ((float4*)y)[i];
    const float4 sc = *(const float4*)(scsh + f);
    const float4 sh = *(const float4*)(scsh + 256 + f);
    v.x = tanhf(v.x * sc.x + sh.x);
    v.y = tanhf(v.y * sc.y + sh.y);
    v.z = tanhf(v.z * sc.z + sh.z);
    v.w = tanhf(v.w * sc.w + sh.w);
    ((float4*)y)[i] = v;
}

// --------------------------------------------------------------------- launch
extern "C" void kernel_launch(void* const* d_in, const int* in_sizes, int n_in,
                              void* d_out, int out_size, void* d_ws, size_t ws_size,
                              hipStream_t stream) {
    const float* x     = (const float*)d_in[0];
    const int*   rows  = (const int*)  d_in[1];
    const int*   cols  = (const int*)  d_in[2];
    const float* ev    = (const float*)d_in[3];
    const float* w1    = (const float*)d_in[4];
    const float* w2    = (const float*)d_in[5];
    const float* gamma = (const float*)d_in[6];
    const float* beta  = (const float*)d_in[7];

    const int nNodes = in_sizes[0] / IN_F;      // 100000
    const int nEdges = in_sizes[1];             // 1.6M
    const size_t nAcc = (size_t)nNodes * IN_F;  // 25.6M floats

    float* ws   = (float*)d_ws;
    float* bufA = ws;                            // SpMM accumulator; later GEMM2 output
    float* H1   = bufA + nAcc;                   // 65536 (K-pair interleaved)
    float* H2   = H1 + 65536;                    // 65536 (K-pair interleaved)
    float* sums = H2 + 65536;                    // 256
    float* sumsq= sums + 256;                    // 256
    float* scsh = sumsq + 256;                   // 512
    float* out1 = (float*)d_out;                 // GEMM1 output lives in d_out

    const dim3 blk(256);

    zero_f32<<<(unsigned)((nAcc + 255) / 256), blk, 0, stream>>>(bufA, nAcc);
    zero_f32<<<2, blk, 0, stream>>>(sums, 512);

    build_hamilton<<<256, blk, 0, stream>>>(w1, H1);
    build_hamilton<<<256, blk, 0, stream>>>(w2, H2);

    const long long spThreads = (long long)nEdges * 64;
    spmm_atomic<<<(unsigned)((spThreads + 255) / 256), blk, 0, stream>>>(
        x, rows, cols, ev, bufA, nEdges);

    const int mTiles = nNodes / 16;              // 6250 (exact)
    const int gemmBlocks = (mTiles * 4 + 7) / 8; // 4 waves per M tile, 8 waves/block
    gemm256_wmma<<<gemmBlocks, blk, 0, stream>>>(bufA, H1, out1, mTiles);

    bn_stats<<<(nNodes + 127) / 128, blk, 0, stream>>>(out1, sums, sumsq, nNodes);
    bn_finalize<<<1, blk, 0, stream>>>(sums, sumsq, gamma, beta, scsh, nNodes);

    const size_t n4 = nAcc / 4;
    bn_tanh_apply<<<(unsigned)((n4 + 255) / 256), blk, 0, stream>>>(out1, scsh, n4);

    gemm256_wmma<<<gemmBlocks, blk, 0, stream>>>(out1, H2, bufA, mTiles);
    (void)hipMemcpyAsync(d_out, bufA, nAcc * sizeof(float), hipMemcpyDeviceToDevice, stream);
}